// CRF_23252952940898
// MI455X (gfx1250) — compile-verified
//
#include <hip/hip_runtime.h>
#include <hip/hip_bf16.h>
#include <math.h>

typedef __attribute__((ext_vector_type(2))) float v2f;
typedef __attribute__((ext_vector_type(8))) float v8f;

#define S_LEN     512
#define B_TOT     256
#define T_TAG     64
#define START_TAG 62
#define STOP_TAG  63
#define VERY_NEG  (-1e30f)
#define BT        16          // batches per block (WMMA M dim)
#define LDSTRIDE  68          // 64 + 4 pad -> conflict-free LDS

// ---------------------------------------------------------------------------
// Kernel 1: gold path score. One block per sequence row i in [0, S]; thread=b.
// Row i contributes iff i <= lengths[b]. Fixed-order tree reduction -> partial.
// ---------------------------------------------------------------------------
__global__ __launch_bounds__(256) void crf_gold_kernel(
    const float* __restrict__ feats, const float* __restrict__ trans,
    const int* __restrict__ target, const int* __restrict__ lengths,
    float* __restrict__ gold_partial) {
  __shared__ float red[256];
  const int i = blockIdx.x;      // 0..S
  const int b = threadIdx.x;     // 0..255
  const int len = lengths[b];
  float val = 0.0f;
  if (i <= len) {
    const int to  = (i < S_LEN) ? target[i * B_TOT + b] : STOP_TAG;
    const int frm = (i == 0) ? START_TAG : target[(i - 1) * B_TOT + b];
    val = trans[to * T_TAG + frm];
    if (i < S_LEN) val += feats[((size_t)i * B_TOT + b) * T_TAG + to];
  }
  red[b] = val;
  __syncthreads();
#pragma unroll
  for (int off = 128; off > 0; off >>= 1) {
    if (b < off) red[b] += red[b + off];
    __syncthreads();
  }
  if (b == 0) gold_partial[i] = red[0];
}

// ---------------------------------------------------------------------------
// Kernel 2: forward algorithm. 16 blocks x 128 threads (4 waves).
// Block owns 16 batches; wave w owns 'to' columns [16w, 16w+16).
// Per step: nxt[b,to] = feat + M[b] + log( sum_frm exp(fv[b,frm]-M[b]) * E[to,frm] )
// with E = exp(trans) held register-resident in WMMA B layout for all 512 steps.
// V_WMMA_F32_16X16X4_F32 layouts (ISA 7.12.2):
//   A 16x4 : lane = row (lo), VGPR0 = K=2*hi, VGPR1 = K=2*hi+1
//   B 4x16 : lane = col (lo), VGPR0 = K=2*hi, VGPR1 = K=2*hi+1
//   D 16x16: VGPR j, lane -> (row = j + 8*hi, col = lo)
// ---------------------------------------------------------------------------
__global__ __launch_bounds__(128) void crf_forward_kernel(
    const float* __restrict__ feats, const float* __restrict__ trans,
    const int* __restrict__ lengths, float* __restrict__ alpha_partial) {
  __shared__ float fv[BT * LDSTRIDE];
  __shared__ float Mbuf[BT];

  const int tid  = threadIdx.x;
  const int wave = tid >> 5;
  const int lane = tid & 31;
  const int hi   = lane >> 4;
  const int lo   = lane & 15;
  const int b0   = blockIdx.x * BT;
  const int to   = wave * 16 + lo;

  // E^T tile (B operand), constant across the whole time loop.
  v2f Btile[16];
#pragma unroll
  for (int kk = 0; kk < 16; ++kk) {
    const int frm = 4 * kk + 2 * hi;
    v2f bv = { __expf(trans[to * T_TAG + frm]),
               __expf(trans[to * T_TAG + frm + 1]) };
    Btile[kk] = bv;
  }

  // Per-lane lengths for the D-layout masked update.
  int lenv[8];
#pragma unroll
  for (int j = 0; j < 8; ++j) lenv[j] = lengths[b0 + j + 8 * hi];

  // fv init: only START is reachable.
  for (int idx = tid; idx < BT * LDSTRIDE; idx += 128) {
    const int k = idx % LDSTRIDE;
    fv[idx] = (k == START_TAG) ? 0.0f : VERY_NEG;
  }
  __syncthreads();

  const float* fbase = feats + (size_t)b0 * T_TAG;

  for (int t = 0; t < S_LEN; ++t) {
    // ---- read fv in A layout; fold in row max ----
    float a[32];
    float m = -3.4e38f;
    const int rowoff = lo * LDSTRIDE + 2 * hi;
#pragma unroll
    for (int kk = 0; kk < 16; ++kk) {
      const float x0 = fv[rowoff + 4 * kk];
      const float x1 = fv[rowoff + 4 * kk + 1];
      a[2 * kk] = x0; a[2 * kk + 1] = x1;
      m = fmaxf(m, fmaxf(x0, x1));
    }
    m = fmaxf(m, __shfl_xor(m, 16, 32));
    if (tid < 16) Mbuf[tid] = m;      // wave 0 publishes row maxes
    __syncthreads();                  // all fv reads done; Mbuf valid

    // Emission tile for this step (issued early to overlap with exp/WMMA).
    const float* ft = fbase + (size_t)t * (B_TOT * T_TAG);
    float femit[8];
#pragma unroll
    for (int j = 0; j < 8; ++j) femit[j] = ft[(j + 8 * hi) * T_TAG + to];

    // P = exp(fv - M)  (VERY_NEG / -inf cleanly map to 0)
#pragma unroll
    for (int v = 0; v < 32; ++v) a[v] = __expf(a[v] - m);

    // ---- GEMM: [16 batch x 64] x E^T[64 x 16] via 16 chained f32 WMMAs ----
    v8f c = {0.f, 0.f, 0.f, 0.f, 0.f, 0.f, 0.f, 0.f};
#pragma unroll
    for (int kk = 0; kk < 16; ++kk) {
      v2f av = { a[2 * kk], a[2 * kk + 1] };
      c = __builtin_amdgcn_wmma_f32_16x16x4_f32(false, av, false, Btile[kk],
                                                (short)0, c, false, false);
    }

    // ---- fv_new = feat + M + log(sum); masked (predicated) store ----
#pragma unroll
    for (int j = 0; j < 8; ++j) {
      const int bl = j + 8 * hi;
      const float nv = femit[j] + Mbuf[bl] + __logf(c[j]);
      if (t < lenv[j]) fv[bl * LDSTRIDE + to] = nv;
    }
    __syncthreads();
  }

  // ---- alpha[b] = logsumexp(fv[b,:] + trans[STOP,:]); fixed-order block sum ----
  if (wave == 0) {
    float asum = 0.0f;
    for (int r = 0; r < BT; ++r) {
      float x0 = fv[r * LDSTRIDE + lane]      + trans[STOP_TAG * T_TAG + lane];
      float x1 = fv[r * LDSTRIDE + lane + 32] + trans[STOP_TAG * T_TAG + lane + 32];
      float mx = fmaxf(x0, x1);
#pragma unroll
      for (int off = 16; off > 0; off >>= 1) mx = fmaxf(mx, __shfl_xor(mx, off, 32));
      float s = __expf(x0 - mx) + __expf(x1 - mx);
#pragma unroll
      for (int off = 16; off > 0; off >>= 1) s += __shfl_xor(s, off, 32);
      asum += mx + __logf(s);
    }
    if (lane == 0) alpha_partial[blockIdx.x] = asum;
  }
}

// ---------------------------------------------------------------------------
// Kernel 3: deterministic final combine: loss = (sum(alpha) - gold) / B
// ---------------------------------------------------------------------------
__global__ void crf_final_kernel(const float* __restrict__ ws,
                                 float* __restrict__ out) {
  if (threadIdx.x == 0 && blockIdx.x == 0) {
    float gold = 0.0f;
    for (int i = 0; i < S_LEN + 1; ++i) gold += ws[i];
    float alpha = 0.0f;
    for (int i = 0; i < B_TOT / BT; ++i) alpha += ws[S_LEN + 1 + i];
    out[0] = (alpha - gold) / (float)B_TOT;
  }
}

extern "C" void kernel_launch(void* const* d_in, const int* in_sizes, int n_in,
                              void* d_out, int out_size, void* d_ws, size_t ws_size,
                              hipStream_t stream) {
  (void)in_sizes; (void)n_in; (void)out_size; (void)ws_size;
  const float* feats   = (const float*)d_in[0];
  const float* trans   = (const float*)d_in[1];
  const int*   target  = (const int*)d_in[2];
  const int*   lengths = (const int*)d_in[3];
  float* out = (float*)d_out;
  float* ws  = (float*)d_ws;   // [0, S+1): gold partials; [S+1, S+1+16): alpha partials

  crf_gold_kernel<<<S_LEN + 1, B_TOT, 0, stream>>>(feats, trans, target, lengths, ws);
  crf_forward_kernel<<<B_TOT / BT, 128, 0, stream>>>(feats, trans, lengths,
                                                     ws + (S_LEN + 1));
  crf_final_kernel<<<1, 1, 0, stream>>>(ws, out);
}